// SpatialGlimpse_37563783971099
// MI455X (gfx1250) — compile-verified
//
#include <hip/hip_runtime.h>
#include <hip/hip_bf16.h>
#include <stdint.h>

// SpatialGlimpse on MI455X (gfx1250).
//
// Memory-bound gather+pool: ~17 MB global reads + 3 MB writes, fully L2-
// resident (X = 64 MB << 192 MB L2) => ~1 us at HBM roofline. WMMA is
// deliberately unused (AI ~0.2 flop/byte; pooling matrices are block-diagonal
// ones => 16x wasted MACs, zero byte savings). The CDNA5-specific win is the
// async global->LDS path: stage each glimpse window once into LDS with
// global_load_async_to_lds_b32 (ASYNCcnt, drained by s_wait_asynccnt), with
// out-of-image elements zero-filled by ds_store (async loads skipped there).
//
// Each 512x512 image's 256x256 depth-2 window (which contains the depth-0/1
// crops) is split into 4 row sub-blocks of 64x256 = 64 KB LDS each:
//  - 64 KB dynamic LDS is safe under any runtime cap (vs 256 KB single-tile),
//  - 256 blocks x 256 threads (8 wave32s) covers the WGP array better,
//  - 4-5 workgroups can co-reside per 320 KB WGP.
// Sub-block s owns tile rows [64s, 64s+64):
//   d2: y in [16s,16s+16)              (rows 4y..4y+3)
//   d1: s=1 -> y in [0,32), s=2 -> y in [32,64)   (rows 64+2y, 64+2y+1)
//   d0: s=1 -> y in [0,32), s=2 -> y in [32,64)   (row  96+y)

#define RETINA   64
#define BATCH    64
#define HH       512
#define WW       512
#define PADC     128
#define TILE     256                    // depth-2 crop side (columns)
#define SROWS    64                     // rows per sub-block
#define STILE_ELEMS (SROWS * TILE)      // 16384 floats = 64 KB
#define OUT_PER_B (3 * RETINA * RETINA) // 12288

extern "C" __global__ void __launch_bounds__(256)
glimpse_kernel(const float* __restrict__ X,
               const float* __restrict__ loc,
               float* __restrict__ out)
{
    extern __shared__ __align__(16) float tile[];   // dynamic LDS, 64 KB

    const int blk  = blockIdx.x;
    const int b    = blk >> 2;          // image index
    const int s    = blk & 3;           // row sub-block 0..3
    const int tid  = threadIdx.x;
    const int nthr = blockDim.x;

    // --- crop center: idx = floor(PAD + (clip(loc,-1,1)+1)*hw/2) (exact in f32)
    float l0 = loc[2 * b + 0];
    float l1 = loc[2 * b + 1];
    l0 = fminf(fmaxf(l0, -1.0f), 1.0f);
    l1 = fminf(fmaxf(l1, -1.0f), 1.0f);
    const int idx0 = (int)floorf((float)PADC + ((l0 + 1.0f) * (float)HH) * 0.5f);
    const int idx1 = (int)floorf((float)PADC + ((l1 + 1.0f) * (float)WW) * 0.5f);

    // depth-2 window origin in ORIGINAL (unpadded) image coords: idx - 256.
    const int ry0 = idx0 - 2 * PADC + SROWS * s;    // this sub-block's first row
    const int cx0 = idx1 - 2 * PADC;

    // --- stage 64x256 sub-window into LDS via CDNA5 async global->LDS loads --
    // AS3->flat cast puts the wave-relative LDS byte offset in addr[31:0],
    // which is exactly what the async VDST VGPR expects.
    const uint32_t lds_base = (uint32_t)(uintptr_t)tile;
    const uint64_t xbase    = (uint64_t)(uintptr_t)(X + (size_t)b * HH * WW);

    for (int e = tid; e < STILE_ELEMS; e += nthr) {
        const int tr = e >> 8;          // local tile row 0..63
        const int tc = e & (TILE - 1);  // tile col 0..255
        const int r  = ry0 + tr;
        const int c  = cx0 + tc;
        const uint32_t lo = lds_base + (uint32_t)(e << 2);
        if ((unsigned)r < (unsigned)HH && (unsigned)c < (unsigned)WW) {
            const uint32_t go = (uint32_t)(((r * WW) + c) << 2);
            // GVS mode: mem = SGPR64 base + per-lane 32-bit offset.
            // Per-lane transfer => EXEC-masked lanes move nothing (zero pad).
            asm volatile("global_load_async_to_lds_b32 %0, %1, %2"
                         :
                         : "v"(lo), "v"(go), "s"(xbase)
                         : "memory");
        } else {
            tile[e] = 0.0f;             // zero padding via ds_store
        }
    }
    asm volatile("s_wait_asynccnt 0" ::: "memory");   // drain this wave's asyncs
    __syncthreads();                                  // publish to all waves

    // --- pooling from LDS -> global (coalesced b32 stores) -------------------
    const size_t ob = (size_t)b * OUT_PER_B;

    // depth 2: 4x4 mean; this sub-block produces y = 16s..16s+15.
    // local rows 4*yy (16B-aligned rows -> ds_load_b128)
    for (int o = tid; o < 16 * RETINA; o += nthr) {
        const int yy = o >> 6, x = o & 63;
        const float* p = &tile[(4 * yy) * TILE + 4 * x];
        float acc = 0.0f;
#pragma unroll
        for (int a = 0; a < 4; ++a) {
            const float4 v = *(const float4*)(p + a * TILE);
            acc += (v.x + v.y) + (v.z + v.w);
        }
        out[ob + 8192 + (16 * s + yy) * RETINA + x] = acc * 0.0625f;
    }

    if (s == 1 || s == 2) {
        const int ybase = 32 * (s - 1);             // 0 (s=1) or 32 (s=2)

        // depth 1: 2x2 mean; rows 64+2y,64+2y+1 -> local rows 2*ly, 2*ly+1
        for (int o = tid; o < 32 * RETINA; o += nthr) {
            const int ly = o >> 6, x = o & 63;
            const float* p = &tile[(2 * ly) * TILE + (64 + 2 * x)];
            const float acc = p[0] + p[1] + p[TILE] + p[TILE + 1];
            out[ob + 4096 + (ybase + ly) * RETINA + x] = acc * 0.25f;
        }

        // depth 0: direct copy; row 96+y -> local row ly+32 (s=1) or ly (s=2)
        const int lr0 = (s == 1) ? 32 : 0;
        for (int o = tid; o < 32 * RETINA; o += nthr) {
            const int ly = o >> 6, x = o & 63;
            out[ob + (ybase + ly) * RETINA + x] = tile[(lr0 + ly) * TILE + (96 + x)];
        }
    }
}

extern "C" void kernel_launch(void* const* d_in, const int* in_sizes, int n_in,
                              void* d_out, int out_size, void* d_ws, size_t ws_size,
                              hipStream_t stream) {
    const float* X   = (const float*)d_in[0];   // [64, 512, 512] f32
    const float* loc = (const float*)d_in[1];   // [64, 2] f32
    float* out       = (float*)d_out;           // [64, 12288] f32

    const size_t shmem = (size_t)STILE_ELEMS * sizeof(float);  // 65536 B
    glimpse_kernel<<<dim3(BATCH * 4), dim3(256), shmem, stream>>>(X, loc, out);
}